// ForceField_30820685316649
// MI455X (gfx1250) — compile-verified
//
#include <hip/hip_runtime.h>
#include <hip/hip_bf16.h>
#include <math.h>

typedef __attribute__((ext_vector_type(2))) float v2f;
typedef __attribute__((ext_vector_type(8))) float v8f;

#define DEG2RAD 0.017453292519943295f
#define TPW 8   // column tiles per wave in the nonbonded kernel

__device__ __forceinline__ void wave_reduce_atomic(float e, float* out) {
    #pragma unroll
    for (int o = 16; o > 0; o >>= 1) e += __shfl_down(e, o, 32);
    if ((threadIdx.x & 31) == 0) unsafeAtomicAdd(out, e);
}

// ---------------------------------------------------------------------------
// init: zero the accumulator
// ---------------------------------------------------------------------------
__global__ void ff_init_kernel(float* out) { out[0] = 0.0f; }

// ---------------------------------------------------------------------------
// Per-tile LJ + Coulomb over the 8 elements this lane owns.
// MASKED=true only for diagonal tiles (strict upper triangle).
// N is a compile-time constant in the specialized instantiation, so r*N
// folds into the global_load 24-bit immediate offset.
// ---------------------------------------------------------------------------
template<bool MASKED>
__device__ __forceinline__ float ff_tile_energy(const v8f& d2,
                                                const float* __restrict__ eps,
                                                const float* __restrict__ rmin,
                                                const float* __restrict__ kqq,
                                                int off, int N, int n, int m0) {
    float acc = 0.0f;
    #pragma unroll
    for (int r = 0; r < 8; ++r) {
        const float ev = eps [off + r * N];   // const ioffset when N constexpr
        const float rv = rmin[off + r * N];
        const float qv = kqq [off + r * N];

        float dd = fmaxf(d2[r], 1e-12f);
        float y  = __builtin_amdgcn_rsqf(dd);         // ~1/sqrt(dd)
        y = y * fmaf(-0.5f * dd * y, y, 1.5f);        // 1 Newton step
        float rod = rv * y;
        float q   = qv;
        if (MASKED) {
            // mask BEFORE the 12th power so the diagonal inf never appears
            const float wt = (n > m0 + r) ? 1.0f : 0.0f;
            rod *= wt;
            q   *= wt;
        }
        float rod2 = rod * rod;
        float rod6 = rod2 * rod2 * rod2;
        acc = fmaf(ev, fmaf(rod6, rod6, -2.0f * rod6), acc);
        acc = fmaf(q, y, acc);
    }
    return acc;
}

// ---------------------------------------------------------------------------
// Nonbonded: one wave per (row-tile, strip of TPW column tiles).
// Per 16x16 tile, d2 computed with one V_WMMA_F32_16X16X4_F32:
//   A = -2 * x[rows] (16x4, K padded), B = x[cols]^T (4x16),
//   C preloaded with |x_m|^2 + |x_n|^2  ==>  D = d2 directly.
// Per-wave partial written to a unique ws slot (no atomic contention).
// NC > 0: compile-time N.  NC == 0: runtime N fallback.
// ---------------------------------------------------------------------------
template<int NC>
__global__ __launch_bounds__(32)
void ff_nonbonded_kernel(const float* __restrict__ x,
                         const float* __restrict__ eps,
                         const float* __restrict__ rmin,
                         const float* __restrict__ kqq,
                         int Nrt, float* __restrict__ ws) {
    const int N = (NC > 0) ? NC : Nrt;

    const int strip = blockIdx.x;                 // column strip of TPW tiles
    const int tr    = blockIdx.y;                 // row tile
    const int W     = tr * gridDim.x + strip;     // unique wave-job id

    const int lane = threadIdx.x;
    const int half = lane >> 4;                   // 0: K=0,1   1: K=2,3
    const int l15  = lane & 15;
    const int rbase = tr * 16;

    // --- A operand (rows of x, pre-scaled by -2): loaded once per wave ----
    const float* xr = x + 3 * (rbase + l15);
    v2f a;
    if (half == 0) { a.x = -2.0f * xr[0]; a.y = -2.0f * xr[1]; }
    else           { a.x = -2.0f * xr[2]; a.y = 0.0f; }

    // --- |x_m|^2 for the 8 rows this lane owns (C/D layout): once per wave
    const int m0 = rbase + 8 * half;
    float sqm[8];
    #pragma unroll
    for (int r = 0; r < 8; ++r) {
        const float* xm = x + 3 * (m0 + r);
        sqm[r] = xm[0] * xm[0] + xm[1] * xm[1] + xm[2] * xm[2];
    }

    float acc = 0.0f;

    #pragma unroll 1
    for (int t = 0; t < TPW; ++t) {
        const int tc = strip * TPW + t;
        if (tc < tr) continue;                    // wave-uniform branch
        const int cbase = tc * 16;
        const int n = cbase + l15;

        // --- B operand + sq[n] for this column tile -----------------------
        const float* xc = x + 3 * (cbase + l15);
        const float cx0 = xc[0], cx1 = xc[1], cx2 = xc[2];
        v2f b;
        if (half == 0) { b.x = cx0; b.y = cx1; }
        else           { b.x = cx2; b.y = 0.0f; }
        const float sqn = cx0 * cx0 + cx1 * cx1 + cx2 * cx2;

        v8f c;
        #pragma unroll
        for (int r = 0; r < 8; ++r) c[r] = sqm[r] + sqn;

        // --- d2 tile in one matrix op -------------------------------------
        v8f d2 = __builtin_amdgcn_wmma_f32_16x16x4_f32(
            false, a, false, b, (short)0, c, false, false);

        // --- element math; mask only needed on diagonal tiles -------------
        const int off = m0 * N + n;               // 32-bit index (N<=4096 ok)
        if (tc > tr) acc += ff_tile_energy<false>(d2, eps, rmin, kqq, off, N, n, m0);
        else         acc += ff_tile_energy<true >(d2, eps, rmin, kqq, off, N, n, m0);
    }

    // --- wave reduce, plain store to unique slot --------------------------
    #pragma unroll
    for (int o = 16; o > 0; o >>= 1) acc += __shfl_down(acc, o, 32);
    if (lane == 0) ws[W] = acc;
}

// ---------------------------------------------------------------------------
// Sum the nonbonded partials into the output (single atomic)
// ---------------------------------------------------------------------------
__global__ __launch_bounds__(256)
void ff_reduce_kernel(const float* __restrict__ ws, int n, float* __restrict__ out) {
    __shared__ float sm[256];
    int t = threadIdx.x;
    float a = 0.0f;
    for (int i = t; i < n; i += 256) a += ws[i];
    sm[t] = a;
    __syncthreads();
    for (int s = 128; s > 0; s >>= 1) {
        if (t < s) sm[t] += sm[t + s];
        __syncthreads();
    }
    if (t == 0) unsafeAtomicAdd(out, sm[0]);
}

// ---------------------------------------------------------------------------
// Harmonic distance terms (bonds, Urey-Bradley)
// ---------------------------------------------------------------------------
__global__ __launch_bounds__(256)
void ff_harm_dist_kernel(const float* __restrict__ x,
                         const int* __restrict__ idx,
                         const float* __restrict__ kk,
                         const float* __restrict__ r0,
                         int M, float* __restrict__ out) {
    int i = blockIdx.x * blockDim.x + threadIdx.x;
    float e = 0.0f;
    if (i < M) {
        int ia = idx[2 * i], ib = idx[2 * i + 1];
        float dx = x[3 * ia]     - x[3 * ib];
        float dy = x[3 * ia + 1] - x[3 * ib + 1];
        float dz = x[3 * ia + 2] - x[3 * ib + 2];
        float d = sqrtf(dx * dx + dy * dy + dz * dz);
        float t = d - r0[i];
        e = kk[i] * t * t;
    }
    wave_reduce_atomic(e, out);
}

// ---------------------------------------------------------------------------
// Angle terms
// ---------------------------------------------------------------------------
__global__ __launch_bounds__(256)
void ff_angle_kernel(const float* __restrict__ x,
                     const int* __restrict__ idx,
                     const float* __restrict__ kt,
                     const float* __restrict__ t0,
                     int M, float* __restrict__ out) {
    int i = blockIdx.x * blockDim.x + threadIdx.x;
    float e = 0.0f;
    if (i < M) {
        int ia = idx[3 * i], ib = idx[3 * i + 1], ic = idx[3 * i + 2];
        float bax = x[3 * ia]     - x[3 * ib];
        float bay = x[3 * ia + 1] - x[3 * ib + 1];
        float baz = x[3 * ia + 2] - x[3 * ib + 2];
        float bcx = x[3 * ic]     - x[3 * ib];
        float bcy = x[3 * ic + 1] - x[3 * ib + 1];
        float bcz = x[3 * ic + 2] - x[3 * ib + 2];
        float dot = bax * bcx + bay * bcy + baz * bcz;
        float la  = sqrtf(bax * bax + bay * bay + baz * baz);
        float lc  = sqrtf(bcx * bcx + bcy * bcy + bcz * bcz);
        float ct  = dot / (la * lc);
        ct = fminf(fmaxf(ct, -0.9999f), 0.9999f);
        float theta = acosf(ct);
        float t = theta - t0[i] * DEG2RAD;
        e = kt[i] * t * t;
    }
    wave_reduce_atomic(e, out);
}

// ---------------------------------------------------------------------------
// Signed torsion angle for a quadruple
// ---------------------------------------------------------------------------
__device__ __forceinline__ float ff_torsion(const float* __restrict__ x,
                                            int i0, int i1, int i2, int i3) {
    float abx = x[3 * i1]     - x[3 * i0];
    float aby = x[3 * i1 + 1] - x[3 * i0 + 1];
    float abz = x[3 * i1 + 2] - x[3 * i0 + 2];
    float bcx = x[3 * i2]     - x[3 * i1];
    float bcy = x[3 * i2 + 1] - x[3 * i1 + 1];
    float bcz = x[3 * i2 + 2] - x[3 * i1 + 2];
    float cdx = x[3 * i3]     - x[3 * i2];
    float cdy = x[3 * i3 + 1] - x[3 * i2 + 1];
    float cdz = x[3 * i3 + 2] - x[3 * i2 + 2];
    float n1x = aby * bcz - abz * bcy;
    float n1y = abz * bcx - abx * bcz;
    float n1z = abx * bcy - aby * bcx;
    float n2x = bcy * cdz - bcz * cdy;
    float n2y = bcz * cdx - bcx * cdz;
    float n2z = bcx * cdy - bcy * cdx;
    float dot = n1x * n2x + n1y * n2y + n1z * n2z;
    float l1 = sqrtf(n1x * n1x + n1y * n1y + n1z * n1z);
    float l2 = sqrtf(n2x * n2x + n2y * n2y + n2z * n2z);
    float cd_ = dot / (l1 * l2);
    cd_ = fminf(fmaxf(cd_, -0.9999f), 0.9999f);
    float ang = acosf(cd_);
    float cxx = n1y * n2z - n1z * n2y;
    float cyy = n1z * n2x - n1x * n2z;
    float czz = n1x * n2y - n1y * n2x;
    float s = cxx * bcx + cyy * bcy + czz * bcz;
    float sg = (s > 0.0f) ? 1.0f : ((s < 0.0f) ? -1.0f : 0.0f);
    return ang * sg;
}

// ---------------------------------------------------------------------------
// Proper dihedrals: kd * (1 + cos(nd*delta - d0))
// ---------------------------------------------------------------------------
__global__ __launch_bounds__(256)
void ff_dihedral_kernel(const float* __restrict__ x,
                        const int* __restrict__ idx,
                        const float* __restrict__ kd,
                        const float* __restrict__ nd,
                        const float* __restrict__ d0,
                        int M, float* __restrict__ out) {
    int i = blockIdx.x * blockDim.x + threadIdx.x;
    float e = 0.0f;
    if (i < M) {
        float delta = ff_torsion(x, idx[4 * i], idx[4 * i + 1], idx[4 * i + 2], idx[4 * i + 3]);
        e = kd[i] * (1.0f + cosf(nd[i] * delta - d0[i] * DEG2RAD));
    }
    wave_reduce_atomic(e, out);
}

// ---------------------------------------------------------------------------
// Impropers: kp * (psi - p0)^2
// ---------------------------------------------------------------------------
__global__ __launch_bounds__(256)
void ff_improper_kernel(const float* __restrict__ x,
                        const int* __restrict__ idx,
                        const float* __restrict__ kp,
                        const float* __restrict__ p0,
                        int M, float* __restrict__ out) {
    int i = blockIdx.x * blockDim.x + threadIdx.x;
    float e = 0.0f;
    if (i < M) {
        float psi = ff_torsion(x, idx[4 * i], idx[4 * i + 1], idx[4 * i + 2], idx[4 * i + 3]);
        float t = psi - p0[i] * DEG2RAD;
        e = kp[i] * t * t;
    }
    wave_reduce_atomic(e, out);
}

// ---------------------------------------------------------------------------
// Central restraint (active only if *central != 0)
// ---------------------------------------------------------------------------
__global__ __launch_bounds__(256)
void ff_central_kernel(const float* __restrict__ x,
                       const int* __restrict__ central,
                       int n, float* __restrict__ out) {
    if (central[0] == 0) return;     // uniform across the block
    __shared__ float sx[256], sy[256], sz[256];
    int t = threadIdx.x;
    float ax = 0.0f, ay = 0.0f, az = 0.0f;
    for (int i = t; i < n; i += 256) {
        ax += x[3 * i]; ay += x[3 * i + 1]; az += x[3 * i + 2];
    }
    sx[t] = ax; sy[t] = ay; sz[t] = az;
    __syncthreads();
    for (int s = 128; s > 0; s >>= 1) {
        if (t < s) { sx[t] += sx[t + s]; sy[t] += sy[t + s]; sz[t] += sz[t + s]; }
        __syncthreads();
    }
    float mx = sx[0] / (float)n, my = sy[0] / (float)n, mz = sz[0] / (float)n;
    __syncthreads();
    float e = 0.0f;
    for (int i = t; i < n; i += 256) {
        float dx = x[3 * i] - mx, dy = x[3 * i + 1] - my, dz = x[3 * i + 2] - mz;
        e += 0.1f * (dx * dx + dy * dy + dz * dz);
    }
    sx[t] = e;
    __syncthreads();
    for (int s = 128; s > 0; s >>= 1) {
        if (t < s) sx[t] += sx[t + s];
        __syncthreads();
    }
    if (t == 0) unsafeAtomicAdd(out, sx[0]);
}

// ---------------------------------------------------------------------------
extern "C" void kernel_launch(void* const* d_in, const int* in_sizes, int n_in,
                              void* d_out, int out_size, void* d_ws, size_t ws_size,
                              hipStream_t stream) {
    const float* x    = (const float*)d_in[0];
    const float* kb   = (const float*)d_in[1];
    const float* b0   = (const float*)d_in[2];
    const float* kt   = (const float*)d_in[3];
    const float* t0   = (const float*)d_in[4];
    const float* ku   = (const float*)d_in[5];
    const float* s0   = (const float*)d_in[6];
    const float* kd   = (const float*)d_in[7];
    const float* nd   = (const float*)d_in[8];
    const float* d0   = (const float*)d_in[9];
    const float* kp   = (const float*)d_in[10];
    const float* p0   = (const float*)d_in[11];
    const float* eps  = (const float*)d_in[12];
    const float* rmin = (const float*)d_in[13];
    const float* kqq  = (const float*)d_in[14];
    const int* bond   = (const int*)d_in[15];
    const int* angle  = (const int*)d_in[16];
    const int* ub     = (const int*)d_in[17];
    const int* dih    = (const int*)d_in[18];
    const int* imp    = (const int*)d_in[19];
    const int* central= (const int*)d_in[20];
    float* out = (float*)d_out;
    float* ws  = (float*)d_ws;

    const int N   = in_sizes[0] / 3;     // 4096
    const int Mb  = in_sizes[15] / 2;    // 6144 bonds
    const int Ma  = in_sizes[16] / 3;    // 8192 angles
    const int Mu  = in_sizes[17] / 2;    // 4096 UB
    const int Md  = in_sizes[18] / 4;    // 12288 dihedrals
    const int Mi  = in_sizes[19] / 4;    // 2048 impropers

    ff_init_kernel<<<1, 1, 0, stream>>>(out);

    ff_harm_dist_kernel<<<(Mb + 255) / 256, 256, 0, stream>>>(x, bond, kb, b0, Mb, out);
    ff_angle_kernel    <<<(Ma + 255) / 256, 256, 0, stream>>>(x, angle, kt, t0, Ma, out);
    ff_harm_dist_kernel<<<(Mu + 255) / 256, 256, 0, stream>>>(x, ub, ku, s0, Mu, out);
    ff_dihedral_kernel <<<(Md + 255) / 256, 256, 0, stream>>>(x, dih, kd, nd, d0, Md, out);
    ff_improper_kernel <<<(Mi + 255) / 256, 256, 0, stream>>>(x, imp, kp, p0, Mi, out);

    const int T = N / 16;                // tiles per side
    const int S = T / TPW;               // column strips
    const int NJOBS = T * S;             // wave jobs -> ws slots
    if (N == 4096) {
        ff_nonbonded_kernel<4096><<<dim3(S, T), 32, 0, stream>>>(x, eps, rmin, kqq, N, ws);
    } else {
        ff_nonbonded_kernel<0>   <<<dim3(S, T), 32, 0, stream>>>(x, eps, rmin, kqq, N, ws);
    }
    ff_reduce_kernel<<<1, 256, 0, stream>>>(ws, NJOBS, out);

    ff_central_kernel<<<1, 256, 0, stream>>>(x, central, N, out);
}